// FilterDetections_80436147520054
// MI455X (gfx1250) — compile-verified
//
#include <hip/hip_runtime.h>
#include <cstdint>
#include <cstddef>

// FilterDetections for MI455X (gfx1250).
// Memory/control-bound selection workload: no matmul structure -> no WMMA.
// CDNA5 features used:
//  * 320KB LDS per WGP: whole 200KB per-class score slice staged in LDS of a
//    single workgroup (impossible on CDNA4's 64KB).
//  * Async global->LDS engine (ASYNCcnt): strided per-lane b32 score gather
//    and b128 candidate-box gather, drained with s_wait_asynccnt.
//  * wave32, 512-thread blocks (16 waves/WGP).
// Stage 1 does a single-pass histogram top-512 select (one read pass + one
// collect pass over the LDS scores instead of a ~26-pass bit binary search),
// a 2048-key bitonic sort, and the 512-round greedy NMS. Stage 2 merges the
// 10 classes per image with an 8192-key bitonic sort and writes outputs.

namespace {
constexpr int B      = 32;
constexpr int N      = 50000;
constexpr int C      = 10;
constexpr int K      = 512;    // PRE_NMS_TOPK
constexpr int MAXDET = 100;
constexpr float NMS_T = 0.5f;
constexpr uint32_t THR_BITS = 0x3C23D70Au;   // __float_as_uint(0.01f)
constexpr uint32_t BIN_BASE = THR_BITS >> 16; // 0x3C23
constexpr int NBINS  = 862;    // covers bits(0.01f)>>16 .. bits(1.0f)>>16
constexpr int HISTSZ = 1024;   // padded pow2 allocation
constexpr int CAP    = 2048;   // collect capacity (sure + threshold-bin)

// d_out layout (floats), reference tuple order:
// boxes[32,100,4] | scores[32,100] | labels[32,100] | rot[32,100,3] | trans[32,100,3]
constexpr int OFF_BOXES  = 0;
constexpr int OFF_SCORES = B * MAXDET * 4;           // 12800
constexpr int OFF_LABELS = OFF_SCORES + B * MAXDET;  // 16000
constexpr int OFF_ROT    = OFF_LABELS + B * MAXDET;  // 19200
constexpr int OFF_TRANS  = OFF_ROT + B * MAXDET * 3; // 28800

// stage1 LDS layout (bytes)
constexpr int SM_SCORES = 0;                          // N*4 = 200000 (pad 200704)
constexpr int SM_KEYS   = 200704;                     // CAP*8 = 16384
constexpr int SM_STAGE  = SM_KEYS + CAP * 8;          // K*16 = 8192 (box staging)
constexpr int SM_X1     = SM_STAGE + K * 16;          // 5 SoA arrays of K floats
constexpr int SM_FLAGS  = SM_X1 + 5 * K * 4;          // sup[K], keep[K]
constexpr int SM_HIST   = SM_FLAGS + 2 * K * 4;       // HISTSZ ints
constexpr int SM_CNT    = SM_HIST + HISTSZ * 4;       // counters
constexpr int SM1_BYTES = SM_CNT + 64;                // ~244 KB (< 320KB/WGP)
}  // namespace

// ---- CDNA5 async global->LDS (per-lane). ASYNCcnt-tracked. ----
__device__ __forceinline__ void async_gl2lds_b32(const void* gptr, const void* lptr) {
  // Low 32 bits of a generic pointer to LDS == wave-relative LDS byte offset.
  uint32_t lds_off = (uint32_t)(uintptr_t)lptr;
  uint64_t gaddr   = (uint64_t)(uintptr_t)gptr;
  asm volatile("global_load_async_to_lds_b32 %0, %1, off"
               :: "v"(lds_off), "v"(gaddr) : "memory");
}
__device__ __forceinline__ void async_gl2lds_b128(const void* gptr, const void* lptr) {
  uint32_t lds_off = (uint32_t)(uintptr_t)lptr;
  uint64_t gaddr   = (uint64_t)(uintptr_t)gptr;
  asm volatile("global_load_async_to_lds_b128 %0, %1, off"
               :: "v"(lds_off), "v"(gaddr) : "memory");
}
__device__ __forceinline__ void wait_async_zero() {
  asm volatile("s_wait_asynccnt 0" ::: "memory");
}

// =====================  Stage 1: per (image,class)  =====================
__global__ __launch_bounds__(512) void fd_stage1(
    const float* __restrict__ boxes, const float* __restrict__ cls,
    uint32_t* __restrict__ ws_bits, int32_t* __restrict__ ws_idx) {
  const int bc   = blockIdx.x;   // 0..B*C-1
  const int b    = bc / C;
  const int c    = bc % C;
  const int tid  = threadIdx.x;
  const int nthr = blockDim.x;

  extern __shared__ unsigned char smem[];
  float*    s_sc    = (float*)(smem + SM_SCORES);
  uint64_t* s_key   = (uint64_t*)(smem + SM_KEYS);
  float*    s_stage = (float*)(smem + SM_STAGE);
  float*    s_x1    = (float*)(smem + SM_X1);
  float*    s_y1    = s_x1 + K;
  float*    s_x2    = s_y1 + K;
  float*    s_y2    = s_x2 + K;
  float*    s_ar    = s_y2 + K;
  int*      s_sup   = (int*)(smem + SM_FLAGS);
  int*      s_keep  = s_sup + K;
  int*      s_hist  = (int*)(smem + SM_HIST);
  int*      s_cnt   = (int*)(smem + SM_CNT);

  // ---- async-stage this class' scores (strided gather, stride C*4 bytes) ----
  const float* gsc = cls + (size_t)b * N * C + c;
  for (int n = tid; n < N; n += nthr)
    async_gl2lds_b32(gsc + (size_t)n * C, s_sc + n);
  for (int h = tid; h < HISTSZ; h += nthr) s_hist[h] = 0;  // overlap with async
  wait_async_zero();
  __syncthreads();

  // ---- single-pass histogram on float bit pattern (order preserving) ----
  for (int n = tid; n < N; n += nthr) {
    uint32_t bits = __float_as_uint(s_sc[n]);
    if (bits > THR_BITS) atomicAdd(&s_hist[(bits >> 16) - BIN_BASE], 1);
  }
  __syncthreads();

  // ---- suffix scan: smallest bin set {bin >= tb} with count >= K ----
  if (tid == 0) {
    int run = 0, tb = 0;
    for (int t = NBINS - 1; t >= 0; --t) {
      run += s_hist[t];
      if (run >= K) { tb = t; break; }
    }
    s_cnt[0] = 0;   // collect counter
    s_cnt[1] = tb;  // threshold bin
  }
  for (int k = tid; k < CAP; k += nthr) s_key[k] = 0ull;
  __syncthreads();
  const uint32_t binLo = (uint32_t)s_cnt[1];

  // ---- collect candidates (sure bins + threshold bin), <= CAP typical ----
  for (int n = tid; n < N; n += nthr) {
    uint32_t bits = __float_as_uint(s_sc[n]);
    if (bits > THR_BITS && ((bits >> 16) - BIN_BASE) >= binLo) {
      int pos = atomicAdd(&s_cnt[0], 1);
      if (pos < CAP)
        s_key[pos] = ((uint64_t)bits << 32) | (uint32_t)~(uint32_t)n;
    }
  }
  __syncthreads();

  // ---- bitonic sort CAP keys, descending (ties -> smaller index first) ----
  for (int kk = 2; kk <= CAP; kk <<= 1) {
    for (int j = kk >> 1; j > 0; j >>= 1) {
      for (int i = tid; i < CAP; i += nthr) {
        int ixj = i ^ j;
        if (ixj > i) {
          bool up = ((i & kk) == 0);
          uint64_t a = s_key[i], bb = s_key[ixj];
          bool sw = up ? (a < bb) : (a > bb);
          if (sw) { s_key[i] = bb; s_key[ixj] = a; }
        }
      }
      __syncthreads();
    }
  }
  // first K sorted keys == exact top-K

  // ---- async-stage candidate boxes (16B rows) into LDS, then SoA ----
  {
    uint64_t key = s_key[tid];
    int idx = key ? (int)~(uint32_t)key : 0;
    async_gl2lds_b128(boxes + ((size_t)b * N + idx) * 4, s_stage + tid * 4);
    wait_async_zero();
  }
  __syncthreads();
  {
    uint64_t key = s_key[tid];
    float x1 = 0.f, y1 = 0.f, x2 = 0.f, y2 = 0.f;
    if (key) {
      x1 = s_stage[tid * 4 + 0];
      y1 = s_stage[tid * 4 + 1];
      x2 = s_stage[tid * 4 + 2];
      y2 = s_stage[tid * 4 + 3];
    }
    s_x1[tid] = x1; s_y1[tid] = y1; s_x2[tid] = x2; s_y2[tid] = y2;
    s_ar[tid] = (x2 - x1) * (y2 - y1);
    s_sup[tid] = 0; s_keep[tid] = 0;
  }
  __syncthreads();

  // ---- greedy NMS over sorted candidates (512-wide IoU per round) ----
  for (int i = 0; i < K; ++i) {
    bool keep_i = (s_key[i] != 0ull) && (s_sup[i] == 0);
    __syncthreads();  // separate reads of s_sup[i] from this round's writes
    if (keep_i) {
      if (tid == 0) s_keep[i] = 1;
      float ix1 = fmaxf(s_x1[i], s_x1[tid]);
      float iy1 = fmaxf(s_y1[i], s_y1[tid]);
      float ix2 = fminf(s_x2[i], s_x2[tid]);
      float iy2 = fminf(s_y2[i], s_y2[tid]);
      float inter = fmaxf(ix2 - ix1, 0.f) * fmaxf(iy2 - iy1, 0.f);
      float uni   = s_ar[i] + s_ar[tid] - inter;
      float iou   = (uni > 0.f) ? inter / uni : 0.f;
      if (iou > NMS_T) s_sup[tid] = 1;   // self-suppresses too (matches ref)
    }
    __syncthreads();
  }

  // ---- cap at MAXDET kept detections (cumsum <= 100) ----
  if (tid == 0) {
    int run = 0;
    for (int i = 0; i < K; ++i)
      if (s_keep[i]) { ++run; if (run > MAXDET) s_keep[i] = 0; }
  }
  __syncthreads();

  // ---- emit (score bits or 0, original index) to workspace ----
  uint64_t key = s_key[tid];
  bool fin = (key != 0ull) && (s_keep[tid] != 0);
  ws_bits[(size_t)bc * K + tid] = fin ? (uint32_t)(key >> 32) : 0u;
  ws_idx [(size_t)bc * K + tid] = key ? (int)~(uint32_t)key : 0;
}

// =====================  Stage 2: per image merge  =====================
__global__ __launch_bounds__(512) void fd_stage2(
    const float* __restrict__ boxes, const float* __restrict__ rot,
    const float* __restrict__ trans, const uint32_t* __restrict__ ws_bits,
    const int32_t* __restrict__ ws_idx, float* __restrict__ out) {
  const int b    = blockIdx.x;
  const int tid  = threadIdx.x;
  const int nthr = blockDim.x;
  constexpr int M   = 8192;    // pow2 >= C*K
  constexpr int TOT = C * K;   // 5120

  extern __shared__ unsigned char smem[];
  uint64_t* s_key = (uint64_t*)smem;   // M * 8 = 64KB

  for (int k = tid; k < M; k += nthr) {
    uint64_t key = 0ull;
    if (k < TOT) {
      uint32_t bits = ws_bits[(size_t)b * TOT + k];
      if (bits) key = ((uint64_t)bits << 32) | (uint32_t)~(uint32_t)k;
    }
    s_key[k] = key;
  }
  __syncthreads();

  for (int kk = 2; kk <= M; kk <<= 1) {
    for (int j = kk >> 1; j > 0; j >>= 1) {
      for (int i = tid; i < M; i += nthr) {
        int ixj = i ^ j;
        if (ixj > i) {
          bool up = ((i & kk) == 0);
          uint64_t a = s_key[i], bb = s_key[ixj];
          bool sw = up ? (a < bb) : (a > bb);
          if (sw) { s_key[i] = bb; s_key[ixj] = a; }
        }
      }
      __syncthreads();
    }
  }

  if (tid < MAXDET) {
    uint64_t key = s_key[tid];
    bool valid = (key != 0ull);
    float score = valid ? __uint_as_float((uint32_t)(key >> 32)) : -1.0f;
    int flat = valid ? (int)~(uint32_t)key : 0;   // c*K + slot
    int c = flat / K, slot = flat % K;
    int idx = valid ? ws_idx[((size_t)b * C + c) * K + slot] : 0;

    const float* bp = boxes + ((size_t)b * N + idx) * 4;
    const float* rp = rot   + ((size_t)b * N + idx) * 3;
    const float* tp = trans + ((size_t)b * N + idx) * 3;

    float* ob = out + OFF_BOXES + ((size_t)b * MAXDET + tid) * 4;
    ob[0] = valid ? bp[0] : -1.0f;
    ob[1] = valid ? bp[1] : -1.0f;
    ob[2] = valid ? bp[2] : -1.0f;
    ob[3] = valid ? bp[3] : -1.0f;

    out[OFF_SCORES + b * MAXDET + tid] = score;
    out[OFF_LABELS + b * MAXDET + tid] = valid ? (float)c : -1.0f;

    float* orr = out + OFF_ROT + ((size_t)b * MAXDET + tid) * 3;
    orr[0] = valid ? rp[0] : -1.0f;
    orr[1] = valid ? rp[1] : -1.0f;
    orr[2] = valid ? rp[2] : -1.0f;

    float* ot = out + OFF_TRANS + ((size_t)b * MAXDET + tid) * 3;
    ot[0] = valid ? tp[0] : -1.0f;
    ot[1] = valid ? tp[1] : -1.0f;
    ot[2] = valid ? tp[2] : -1.0f;
  }
}

extern "C" void kernel_launch(void* const* d_in, const int* in_sizes, int n_in,
                              void* d_out, int out_size, void* d_ws, size_t ws_size,
                              hipStream_t stream) {
  (void)in_sizes; (void)n_in; (void)out_size; (void)ws_size;
  const float* boxes = (const float*)d_in[0];
  const float* cls   = (const float*)d_in[1];
  const float* rot   = (const float*)d_in[2];
  const float* trans = (const float*)d_in[3];
  float* out = (float*)d_out;

  // workspace: uint32 score-bits [B*C*K] then int32 indices [B*C*K] (~1.3 MB)
  uint32_t* ws_bits = (uint32_t*)d_ws;
  int32_t*  ws_idx  = (int32_t*)((char*)d_ws + (size_t)B * C * K * sizeof(uint32_t));

  fd_stage1<<<dim3(B * C), dim3(512), SM1_BYTES, stream>>>(boxes, cls, ws_bits, ws_idx);
  fd_stage2<<<dim3(B), dim3(512), 8192 * sizeof(uint64_t), stream>>>(
      boxes, rot, trans, ws_bits, ws_idx, out);
}